// Answer_68616397521538
// MI455X (gfx1250) — compile-verified
//
#include <hip/hip_runtime.h>
#include <math.h>

#define B_   32
#define L_   4096
#define MEM_ 256
#define Q_   256
#define BL_  (B_ * L_)      // 131072
#define BQ_  (B_ * Q_)      // 8192
#define B3Q_ (B_ * 3 * Q_)  // 24576

// workspace layout (float offsets)
#define OFF_S1   0
#define OFF_S2   (OFF_S1 + BL_)
#define OFF_SACC (OFF_S2 + BL_)
#define OFF_EACC (OFF_SACC + BL_)
#define OFF_QB   (OFF_EACC + BL_)
#define OFF_QA   (OFF_QB + BQ_)
#define OFF_QE   (OFF_QA + BQ_)
#define OFF_EV   (OFF_QE + BQ_)
#define OFF_X1   (OFF_EV + BQ_)
#define OFF_S_A  (OFF_X1 + BQ_)
#define OFF_S_B  (OFF_S_A + BQ_)
#define OFF_GI   (OFF_S_B + BQ_)
#define OFF_GH   (OFF_GI + B3Q_)

typedef __attribute__((ext_vector_type(2))) float v2f;
typedef __attribute__((ext_vector_type(8))) float v8f;

// ---------------------------------------------------------------------------
// Generic small GEMM with f32 WMMA:  C[M x N] (+)= A[M x K] @ W[N x K]^T (+bias)
// One wave per 16x16 output tile. Layouts per CDNA5 ISA 7.12.2:
//   A 16x4: lane(0..15)=row m, VGPR0/1 = K{0,1}; lanes 16..31 carry K{2,3}.
//   B 4x16: VGPR v holds row K=v (lanes 0-15, n=lane) and K=v+2 (lanes 16-31).
//   C/D 16x16: VGPR j = row m=j (lanes 0-15, n=lane) / m=j+8 (lanes 16-31).
// ---------------------------------------------------------------------------
__global__ void wmma_gemm_kernel(const float* __restrict__ A, int lda,
                                 const float* __restrict__ W, int ldw,
                                 const float* __restrict__ bias,
                                 float* __restrict__ C, int ldc,
                                 int K, int accumulate) {
    const int lane = threadIdx.x & 31;
    const int row  = lane & 15;
    const int hi   = lane >> 4;          // 0 -> K pair {0,1}, 1 -> {2,3}
    const int m0   = blockIdx.y * 16;
    const int n0   = blockIdx.x * 16;

    const float binit = bias ? bias[n0 + row] : 0.0f;
    v8f acc = {};
    if (accumulate) {
#pragma unroll
        for (int j = 0; j < 8; ++j)
            acc[j] = C[(size_t)(m0 + j + 8 * hi) * ldc + n0 + row] + binit;
    } else {
#pragma unroll
        for (int j = 0; j < 8; ++j) acc[j] = binit;
    }

    const float* aRow = A + (size_t)(m0 + row) * lda + 2 * hi;
    const float* wRow = W + (size_t)(n0 + row) * ldw + 2 * hi;
    for (int k = 0; k < K; k += 4) {
        v2f a; a.x = aRow[k]; a.y = aRow[k + 1];
        v2f b; b.x = wRow[k]; b.y = wRow[k + 1];
        acc = __builtin_amdgcn_wmma_f32_16x16x4_f32(false, a, false, b,
                                                    (short)0, acc, false, false);
    }
#pragma unroll
    for (int j = 0; j < 8; ++j)
        C[(size_t)(m0 + j + 8 * hi) * ldc + n0 + row] = acc[j];
}

// ---------------------------------------------------------------------------
// Fused dual score pass: S1[b,l] = sum_m M[b,l,m]*q1[b,m]
//                        S2[b,l] = sum_m M[b,l,m]*q2[b,m]
// A = M tile (16 l x 4 m), B columns 0/1 = q1/q2 chunks, rest zero.
// One wave per (b, 16-l tile); single read of M produces both score vectors.
// ---------------------------------------------------------------------------
__global__ void scores_kernel(const float* __restrict__ M,
                              const float* __restrict__ q1,
                              const float* __restrict__ q2,
                              float* __restrict__ S1,
                              float* __restrict__ S2) {
    const int lane = threadIdx.x & 31;
    const int row  = lane & 15;          // A row (l in tile) == B column n
    const int hi   = lane >> 4;
    const int b    = blockIdx.y;
    const int l0   = blockIdx.x * 16;

    const float* Mb = M + ((size_t)b * L_ + l0 + row) * MEM_ + 2 * hi;
    const float* qv = ((row == 0) ? q1 : q2) + (size_t)b * MEM_ + 2 * hi;

    __builtin_prefetch(Mb + 16 * MEM_, 0, 0);   // speculative prefetch: next l-tile

    v8f acc = {};
    for (int k = 0; k < MEM_; k += 4) {
        v2f a; a.x = Mb[k]; a.y = Mb[k + 1];
        v2f bb;
        if (row < 2) { bb.x = qv[k]; bb.y = qv[k + 1]; }
        else         { bb.x = 0.0f;  bb.y = 0.0f; }
        acc = __builtin_amdgcn_wmma_f32_16x16x4_f32(false, a, false, bb,
                                                    (short)0, acc, false, false);
    }
    if (row < 2) {
        float* dst = (row == 0) ? S1 : S2;
#pragma unroll
        for (int j = 0; j < 8; ++j)
            dst[(size_t)b * L_ + l0 + j + 8 * hi] = acc[j];
    }
}

// ---------------------------------------------------------------------------
// Fused dual weighted sum: ev[b,m] = sum_l p[b,l]*M[b,l,m]
//                          x1[b,m] = sum_l w[b,l]*M[b,l,m]
// A rows 0/1 = p/w chunks (rest zero), B = M tile (4 l x 16 m).
// One wave per (b, 16-m tile); single read of M produces both vectors.
// ---------------------------------------------------------------------------
__global__ void wsum_kernel(const float* __restrict__ M,
                            const float* __restrict__ p,
                            const float* __restrict__ w,
                            float* __restrict__ ev,
                            float* __restrict__ x1) {
    const int lane = threadIdx.x & 31;
    const int row  = lane & 15;          // A row (0=p, 1=w) == B column n (m in tile)
    const int hi   = lane >> 4;
    const int b    = blockIdx.y;
    const int m0   = blockIdx.x * 16;

    const float* pv = ((row == 0) ? p : w) + (size_t)b * L_ + 2 * hi;
    const float* Mb = M + ((size_t)b * L_ + 2 * hi) * MEM_ + m0 + row;

    v8f acc = {};
    for (int l = 0; l < L_; l += 4) {
        v2f a;
        if (row < 2) { a.x = pv[l]; a.y = pv[l + 1]; }
        else         { a.x = 0.0f;  a.y = 0.0f; }
        v2f bb;
        bb.x = Mb[(size_t)l * MEM_];
        bb.y = Mb[(size_t)l * MEM_ + MEM_];
        acc = __builtin_amdgcn_wmma_f32_16x16x4_f32(false, a, false, bb,
                                                    (short)0, acc, false, false);
    }
    if (hi == 0) {   // D rows 0 and 1 live in VGPR 0/1 of lanes 0..15
        ev[(size_t)b * MEM_ + m0 + row] = acc[0];
        x1[(size_t)b * MEM_ + m0 + row] = acc[1];
    }
}

// ---------------------------------------------------------------------------
// In-place softmax over L per batch row; optionally accumulate scale*p into acc.
// ---------------------------------------------------------------------------
__global__ void softmax_kernel(float* __restrict__ S,
                               float* __restrict__ accDst,
                               float scale) {
    const int b = blockIdx.x;
    const int tid = threadIdx.x;
    __shared__ float red[256];
    float* rowp = S + (size_t)b * L_;

    float m = -3.402823466e38f;
    for (int i = tid; i < L_; i += 256) m = fmaxf(m, rowp[i]);
    red[tid] = m; __syncthreads();
    for (int s = 128; s > 0; s >>= 1) {
        if (tid < s) red[tid] = fmaxf(red[tid], red[tid + s]);
        __syncthreads();
    }
    m = red[0]; __syncthreads();

    float sum = 0.0f;
    for (int i = tid; i < L_; i += 256) {
        float e = __expf(rowp[i] - m);
        rowp[i] = e;
        sum += e;
    }
    red[tid] = sum; __syncthreads();
    for (int s = 128; s > 0; s >>= 1) {
        if (tid < s) red[tid] += red[tid + s];
        __syncthreads();
    }
    const float inv = 1.0f / red[0];

    for (int i = tid; i < L_; i += 256) {
        float pv = rowp[i] * inv;
        rowp[i] = pv;
        if (accDst) accDst[(size_t)b * L_ + i] += scale * pv;
    }
}

// ---------------------------------------------------------------------------
// GRU cell (torch gate order r,z,n): sNew = (1-z)*n + z*sOld
// ---------------------------------------------------------------------------
__global__ void gru_kernel(const float* __restrict__ gi,
                           const float* __restrict__ gh,
                           const float* __restrict__ sOld,
                           float* __restrict__ sNew) {
    int i = blockIdx.x * 256 + threadIdx.x;
    if (i >= BQ_) return;
    int b = i / Q_, q = i % Q_;
    const float* gib = gi + (size_t)b * 3 * Q_;
    const float* ghb = gh + (size_t)b * 3 * Q_;
    float ir = gib[q], iz = gib[Q_ + q], in_ = gib[2 * Q_ + q];
    float hr = ghb[q], hz = ghb[Q_ + q], hn  = ghb[2 * Q_ + q];
    float r = 1.0f / (1.0f + __expf(-(ir + hr)));
    float z = 1.0f / (1.0f + __expf(-(iz + hz)));
    float n = tanhf(in_ + r * hn);
    sNew[i] = (1.0f - z) * n + z * sOld[i];
}

__global__ void init_kernel(const float* __restrict__ s0, float* __restrict__ ws) {
    int i = blockIdx.x * 256 + threadIdx.x;
    if (i < 2 * BL_) ws[OFF_SACC + i] = 0.0f;     // zero both prob accumulators
    if (i < BQ_)     ws[OFF_S_A + i] = s0[i];     // s = s0
}

__global__ void final_kernel(const float* __restrict__ ws, float* __restrict__ out) {
    int i = blockIdx.x * 256 + threadIdx.x;
    if (i < BL_) {
        out[i]       = logf(ws[OFF_SACC + i]);
        out[BL_ + i] = logf(ws[OFF_EACC + i]);
    }
}

extern "C" void kernel_launch(void* const* d_in, const int* in_sizes, int n_in,
                              void* d_out, int out_size, void* d_ws, size_t ws_size,
                              hipStream_t stream) {
    const float* M    = (const float*)d_in[0];
    const float* s0   = (const float*)d_in[1];
    const float* Wb   = (const float*)d_in[2];
    const float* We   = (const float*)d_in[3];
    const float* Wa   = (const float*)d_in[4];
    const float* W_ih = (const float*)d_in[5];
    const float* W_hh = (const float*)d_in[6];
    const float* b_ih = (const float*)d_in[7];
    const float* b_hh = (const float*)d_in[8];

    float* ws   = (float*)d_ws;
    float* S1   = ws + OFF_S1;
    float* S2   = ws + OFF_S2;
    float* sacc = ws + OFF_SACC;
    float* eacc = ws + OFF_EACC;
    float* qb   = ws + OFF_QB;
    float* qa   = ws + OFF_QA;
    float* qe   = ws + OFF_QE;
    float* ev   = ws + OFF_EV;
    float* x1   = ws + OFF_X1;
    float* sA   = ws + OFF_S_A;
    float* sB   = ws + OFF_S_B;
    float* gi   = ws + OFF_GI;
    float* gh   = ws + OFF_GH;

    init_kernel<<<(2 * BL_) / 256, 256, 0, stream>>>(s0, ws);

    auto gemm = [&](const float* A, int lda, const float* W, int ldw,
                    const float* bias, float* C, int ldc, int K, int N, int accF) {
        dim3 g(N / 16, B_ / 16);
        wmma_gemm_kernel<<<g, 32, 0, stream>>>(A, lda, W, ldw, bias, C, ldc, K, accF);
    };

    for (int t = 0; t < 5; ++t) {
        const float* sCur = (t & 1) ? sB : sA;
        float*       sNxt = (t & 1) ? sA : sB;

        // qb = s @ Wb^T ; qa = s @ Wa^T
        gemm(sCur, Q_, Wb, Q_, nullptr, qb, MEM_, Q_, MEM_, 0);
        gemm(sCur, Q_, Wa, Q_, nullptr, qa, MEM_, Q_, MEM_, 0);

        // pass 1 over M: start scores + attention scores (fused)
        scores_kernel<<<dim3(L_ / 16, B_), 32, 0, stream>>>(M, qb, qa, S1, S2);

        // softmaxes: S1 -> p_start (accumulate mean), S2 -> beta
        softmax_kernel<<<B_, 256, 0, stream>>>(S1, sacc, 0.2f);
        softmax_kernel<<<B_, 256, 0, stream>>>(S2, nullptr, 0.0f);

        // pass 2 over M: end_vec = p_start^T M ; x1 = beta^T M (fused)
        wsum_kernel<<<dim3(MEM_ / 16, B_), 32, 0, stream>>>(M, S1, S2, ev, x1);

        // qe = [s, end_vec] @ We^T  (two accumulating GEMMs over We column slices)
        gemm(sCur, Q_,  We,      Q_ + MEM_, nullptr, qe, MEM_, Q_,   MEM_, 0);
        gemm(ev,  MEM_, We + Q_, Q_ + MEM_, nullptr, qe, MEM_, MEM_, MEM_, 1);

        // pass 3 over M: end scores (S2 is a dead duplicate)
        scores_kernel<<<dim3(L_ / 16, B_), 32, 0, stream>>>(M, qe, qe, S1, S2);
        softmax_kernel<<<B_, 256, 0, stream>>>(S1, eacc, 0.2f);

        // GRU: gi = x1 @ W_ih^T + b_ih ; gh = s @ W_hh^T + b_hh
        gemm(x1,  MEM_, W_ih, MEM_, b_ih, gi, 3 * Q_, MEM_, 3 * Q_, 0);
        gemm(sCur, Q_,  W_hh, Q_,   b_hh, gh, 3 * Q_, Q_,   3 * Q_, 0);
        gru_kernel<<<BQ_ / 256, 256, 0, stream>>>(gi, gh, sCur, sNxt);
    }

    final_kernel<<<BL_ / 256, 256, 0, stream>>>(ws, (float*)d_out);
}